// Kalman_31147102830760
// MI455X (gfx1250) — compile-verified
//
#include <hip/hip_runtime.h>

typedef __attribute__((ext_vector_type(2))) float v2f;
typedef __attribute__((ext_vector_type(4))) float v4f;
typedef __attribute__((ext_vector_type(8))) float v8f;

#define T_STEPS 4096
#define D 64
#define DD 4096      // 64*64
#define DD4 1024     // DD/4
#define NTHREADS 512 // 16 waves of 32 -> 4x4 grid of 16x16 WMMA tiles

__device__ __forceinline__ float ldm(const float* M, int r, int c, int t) {
  return t ? M[c * D + r] : M[r * D + c];
}

// Dst = add + alpha * op(A) @ op(B), all 64x64 row-major (LDS or global via
// generic pointers). One 16x16 output tile per wave; K-loop of 16
// V_WMMA_F32_16X16X4_F32. Entry/exit barriers make data deps explicit.
// Caller guarantees Dst does not alias A or B.
__device__ __forceinline__ void mm64(const float* A, int ta,
                                     const float* B, int tb,
                                     float* Dst, const float* add,
                                     float alpha, int tid) {
  __syncthreads();
  const int w = tid >> 5, lane = tid & 31;
  const int ti = (w >> 2) << 4;  // tile row base
  const int tj = (w & 3) << 4;   // tile col base
  const int lo = lane & 15, hi = lane >> 4;
  v8f acc = {0.f, 0.f, 0.f, 0.f, 0.f, 0.f, 0.f, 0.f};
#pragma unroll
  for (int kk = 0; kk < 16; ++kk) {
    const int k0 = kk << 2;
    v2f av, bv;
    // A fragment 16x4: lanes 0-15 -> K=k0+0,k0+1 ; lanes 16-31 -> K=k0+2,k0+3
    av.x = ldm(A, ti + lo, k0 + 2 * hi + 0, ta);
    av.y = ldm(A, ti + lo, k0 + 2 * hi + 1, ta);
    // B fragment 4x16 mirrored layout
    bv.x = ldm(B, k0 + 2 * hi + 0, tj + lo, tb);
    bv.y = ldm(B, k0 + 2 * hi + 1, tj + lo, tb);
    acc = __builtin_amdgcn_wmma_f32_16x16x4_f32(false, av, false, bv,
                                                (short)0, acc, false, false);
  }
#pragma unroll
  for (int i = 0; i < 8; ++i) {
    const int m = ti + i + hi * 8, n = tj + lo;
    float v = alpha * acc[i];
    if (add) v += add[m * D + n];
    Dst[m * D + n] = v;
  }
  __syncthreads();
}

// ---- b128 matrix movers -------------------------------------------------
__device__ __forceinline__ void copy_mat4(float* dst, const float* src,
                                          int tid) {
  v4f* d4 = (v4f*)dst;
  const v4f* s4 = (const v4f*)src;
  for (int i = tid; i < DD4; i += NTHREADS) d4[i] = s4[i];
}

// write-once outputs: non-temporal stores so 67MB of output doesn't evict
// the 130MB ws working set from the 192MB L2 (backward pass reloads ws).
__device__ __forceinline__ void store_mat4_nt(float* __restrict__ dst,
                                              const float* src, int tid) {
  v4f* d4 = (v4f*)dst;
  const v4f* s4 = (const v4f*)src;
  for (int i = tid; i < DD4; i += NTHREADS)
    __builtin_nontemporal_store(s4[i], d4 + i);
}

// prefetch a 16KB matrix region into cache (global_prefetch_b8, one
// cacheline per lane-address; 16B lane stride covers the region in 2 iters)
__device__ __forceinline__ void prefetch_mat(const float* src, int tid) {
  for (int i = tid * 4; i < DD; i += NTHREADS * 4)
    __builtin_prefetch(&src[i], 0, 3);
}

// In-place Cholesky (lower) of SPD 64x64 in LDS. Full trailing-square
// symmetric rank-1 updates keep the buffer symmetric; only lower+diag used.
__device__ void chol64(float* S, int tid) {
  for (int j = 0; j < D; ++j) {
    __syncthreads();
    if (tid == 0) S[j * D + j] = sqrtf(S[j * D + j]);
    __syncthreads();
    {
      int i = j + 1 + tid;
      if (tid < D && i < D) S[i * D + j] /= S[j * D + j];
    }
    __syncthreads();
    const int n = D - 1 - j;
    for (int idx = tid; idx < n * n; idx += NTHREADS) {
      const int ii = j + 1 + idx / n;
      const int cc = j + 1 + idx % n;
      S[ii * D + cc] -= S[ii * D + j] * S[cc * D + j];
    }
  }
  __syncthreads();
}

// W = L^{-1} (strictly lower + diag), L in lower triangle of Lm.
__device__ void trinv64(const float* Lm, float* W, int tid) {
  __syncthreads();
  for (int i = tid; i < DD; i += NTHREADS)
    W[i] = (i / D == i % D) ? 1.0f : 0.0f;
  __syncthreads();
  for (int i = 0; i < D; ++i) {
    if (tid < D) W[i * D + tid] /= Lm[i * D + i];
    __syncthreads();
    const int nrows = D - 1 - i;
    for (int idx = tid; idx < nrows * D; idx += NTHREADS) {
      const int k = i + 1 + idx / D;
      const int c = idx % D;
      W[k * D + c] -= Lm[k * D + i] * W[i * D + c];
    }
    __syncthreads();
  }
}

__global__ __launch_bounds__(NTHREADS) void kalman_scan_kernel(
    const float* __restrict__ gy, const float* __restrict__ gA,
    const float* __restrict__ gb, const float* __restrict__ gQ,
    const float* __restrict__ gH, const float* __restrict__ gc,
    const float* __restrict__ gR, const float* __restrict__ gm0,
    const float* __restrict__ gP0, float* __restrict__ out_means,
    float* __restrict__ out_covs, float* __restrict__ out_ll,
    float* __restrict__ ws_pm, float* __restrict__ ws_pP,
    float* __restrict__ ws_fm, float* __restrict__ ws_fP) {
  __shared__ __align__(16) float sA[DD], sH[DD], Pb[DD], pPb[DD];
  __shared__ __align__(16) float T1[DD], T2[DD], Sm[DD], Xm[DD];
  __shared__ float vm[D], vpm[D], vr[D], vz[D], vsm[D], vd[D];
  __shared__ float llacc;
  const int tid = threadIdx.x;
  const float cLog2Pi = 1.8378770664093453f;

  copy_mat4(sA, gA, tid);
  copy_mat4(sH, gH, tid);
  if (tid == 0) llacc = 0.0f;
  __syncthreads();

  // ======================= forward filter =======================
  for (int t = 0; t < T_STEPS; ++t) {
    if (t == 0) {
      if (tid < D) vpm[tid] = gm0[tid];
      copy_mat4(pPb, gP0, tid);
      __syncthreads();
    } else {
      if (tid < D) {  // pm = A @ m + b
        float s = gb[tid];
        for (int k = 0; k < D; ++k) s += sA[tid * D + k] * vm[k];
        vpm[tid] = s;
      }
      // pP = A P A^T + Q
      mm64(sA, 0, Pb, 0, T2, nullptr, 1.0f, tid);
      mm64(T2, 0, sA, 1, pPb, gQ, 1.0f, tid);
    }
    // stash predicted state for the smoother (stays L2-resident: 130MB<192MB)
    if (tid < D) ws_pm[(size_t)t * D + tid] = vpm[tid];
    copy_mat4(ws_pP + (size_t)t * DD, pPb, tid);

    mm64(sH, 0, pPb, 0, T1, nullptr, 1.0f, tid);   // T1 = H pP
    mm64(T1, 0, sH, 1, Sm, gR, 1.0f, tid);         // S = T1 H^T + R
    chol64(Sm, tid);                               // S -> L (lower)
    trinv64(Sm, T2, tid);                          // T2 = L^{-1}

    if (tid < D) {  // r = y - (H pm + c)
      float s = gc[tid];
      for (int k = 0; k < D; ++k) s += sH[tid * D + k] * vpm[k];
      vr[tid] = gy[(size_t)t * D + tid] - s;
    }
    __syncthreads();
    if (tid < D) {  // z = Linv @ r
      float s = 0.f;
      for (int k = 0; k < D; ++k) s += T2[tid * D + k] * vr[k];
      vz[tid] = s;
    }
    // X = S^{-1} H pP = Linv^T (Linv T1)
    mm64(T2, 0, T1, 0, Sm, nullptr, 1.0f, tid);    // Sm = Linv T1 (L consumed)
    mm64(T2, 1, Sm, 0, Xm, nullptr, 1.0f, tid);    // Xm = Linv^T Sm
    if (tid == 0) {  // log-likelihood term: log Linv_ii = -log L_ii
      float s = -0.5f * (float)D * cLog2Pi;
      for (int i = 0; i < D; ++i) {
        s += logf(T2[i * D + i]);
        s -= 0.5f * vz[i] * vz[i];
      }
      llacc += s;
    }
    if (tid < D) {  // fm = pm + K r = pm + X^T r
      float s = vpm[tid];
      for (int k = 0; k < D; ++k) s += Xm[k * D + tid] * vr[k];
      vm[tid] = s;
    }
    mm64(Xm, 1, T1, 0, Pb, pPb, -1.0f, tid);       // fP = pP - X^T T1
    if (tid < D) ws_fm[(size_t)t * D + tid] = vm[tid];
    copy_mat4(ws_fP + (size_t)t * DD, Pb, tid);
    __syncthreads();
  }

  // ======================= RTS smoother =======================
  if (tid < D) {
    vsm[tid] = vm[tid];
    __builtin_nontemporal_store(vm[tid],
                                &out_means[(size_t)(T_STEPS - 1) * D + tid]);
  }
  store_mat4_nt(out_covs + (size_t)(T_STEPS - 1) * DD, Pb, tid);
  __syncthreads();

  for (int t = T_STEPS - 2; t >= 0; --t) {
    // reload this step's forward state (from L2)...
    copy_mat4(T1, ws_fP + (size_t)t * DD, tid);          // fP_t
    copy_mat4(pPb, ws_pP + (size_t)(t + 1) * DD, tid);   // pP_{t+1}
    // ...and prefetch next iteration's while this step computes
    if (t > 0) {
      prefetch_mat(ws_fP + (size_t)(t - 1) * DD, tid);
      prefetch_mat(ws_pP + (size_t)t * DD, tid);
    }
    if (tid < D) {
      vm[tid] = ws_fm[(size_t)t * D + tid];       // fm_t
      vpm[tid] = ws_pm[(size_t)(t + 1) * D + tid];
      vd[tid] = vsm[tid] - vpm[tid];
    }
    __syncthreads();
    mm64(sA, 0, T1, 0, Xm, nullptr, 1.0f, tid);   // G = A fP_t
    copy_mat4(Sm, pPb, tid);
    chol64(Sm, tid);
    trinv64(Sm, T2, tid);                          // T2 = Linv(pP_{t+1})
    mm64(T2, 0, Xm, 0, sH, nullptr, 1.0f, tid);    // sH scratch = Linv G
    mm64(T2, 1, sH, 0, Xm, nullptr, 1.0f, tid);    // Xm = pP^{-1} G ; C = Xm^T
    for (int i = tid; i < DD; i += NTHREADS) Sm[i] = Pb[i] - pPb[i];  // Δ
    mm64(Xm, 1, Sm, 0, sH, nullptr, 1.0f, tid);    // U = C Δ
    mm64(sH, 0, Xm, 0, Pb, T1, 1.0f, tid);         // sP = fP_t + U C^T
    if (tid < D) {  // sm = fm_t + C (sm_{t+1} - pm_{t+1})
      float s = vm[tid];
      for (int k = 0; k < D; ++k) s += Xm[k * D + tid] * vd[k];
      vsm[tid] = s;
    }
    __syncthreads();
    if (tid < D)
      __builtin_nontemporal_store(vsm[tid], &out_means[(size_t)t * D + tid]);
    store_mat4_nt(out_covs + (size_t)t * DD, Pb, tid);
    __syncthreads();
  }
  if (tid == 0) *out_ll = llacc;
}

extern "C" void kernel_launch(void* const* d_in, const int* in_sizes, int n_in,
                              void* d_out, int out_size, void* d_ws,
                              size_t ws_size, hipStream_t stream) {
  const float* gy = (const float*)d_in[0];
  const float* gA = (const float*)d_in[1];
  const float* gb = (const float*)d_in[2];
  const float* gQ = (const float*)d_in[3];
  const float* gH = (const float*)d_in[4];
  const float* gc = (const float*)d_in[5];
  const float* gR = (const float*)d_in[6];
  const float* gm0 = (const float*)d_in[7];
  const float* gP0 = (const float*)d_in[8];

  float* out = (float*)d_out;
  float* out_means = out;                                   // [T,64]
  float* out_covs = out + (size_t)T_STEPS * D;              // [T,64,64]
  float* out_ll = out_covs + (size_t)T_STEPS * DD;          // scalar

  float* ws = (float*)d_ws;
  float* ws_pm = ws;
  float* ws_pP = ws_pm + (size_t)T_STEPS * D;
  float* ws_fm = ws_pP + (size_t)T_STEPS * DD;
  float* ws_fP = ws_fm + (size_t)T_STEPS * D;

  hipLaunchKernelGGL(kalman_scan_kernel, dim3(1), dim3(NTHREADS), 0, stream,
                     gy, gA, gb, gQ, gH, gc, gR, gm0, gP0, out_means, out_covs,
                     out_ll, ws_pm, ws_pP, ws_fm, ws_fP);
}